// ConvThreshold_90108413870322
// MI455X (gfx1250) — compile-verified
//
#include <hip/hip_runtime.h>

// Scale-adaptive 5x5 Gaussian blur + threshold mask (bandwidth-bound; ~38MB
// traffic -> ~1.6us floor at 23.3 TB/s). WMMA inapplicable (per-pixel varying
// weights => weighted stencil, not GEMM). CDNA5 paths used instead:
//   - TDM tensor_load_to_lds: one 2D-tile DMA per interior workgroup (TENSORcnt)
//   - per-lane global_load_async_to_lds_b32 fallback for left/top border tiles
//   - hardware v_exp_f32 with t^r2 factorization: 1 exp + 4 muls per pixel

#define W_IMG 768
#define H_IMG 768
#define NPIX  (W_IMG * H_IMG)
#define HALO  2
#define TW    64
#define TH    16
#define LW    (TW + 2 * HALO)   // 68 floats per LDS row
#define LH    (TH + 2 * HALO)   // 20 rows
#define NTHREADS 256

typedef unsigned int u32;
typedef unsigned long long u64;
typedef u32 u32x4 __attribute__((ext_vector_type(4)));
typedef u32 u32x8 __attribute__((ext_vector_type(8)));

#define T(dy, dx) tile[(ly + HALO + (dy)) * LW + (lx + HALO + (dx))]

__global__ __launch_bounds__(NTHREADS) void ConvThreshold_90108413870322_kernel(
    const float* __restrict__ bev_map,
    const float* __restrict__ bev_scale,
    float* __restrict__ out_conv,
    float* __restrict__ out_mask)
{
    __shared__ float tile[LH * LW];

    const int tid = threadIdx.x;
    const int n   = blockIdx.z;
    const int x0  = blockIdx.x * TW;
    const int y0  = blockIdx.y * TH;

    const float* img = bev_map  + (size_t)n * NPIX;
    const float* scl = bev_scale + (size_t)n * NPIX;

    // Tiles whose halo does not cross the LEFT/TOP image edge can be fetched
    // with a single TDM 2D-tile DMA: right/bottom overhang is zero-filled by
    // the D# OOB rule (reads beyond tensor_dim return zero).
    const bool tdm_ok = (x0 >= HALO) && (y0 >= HALO);

    if (tdm_ok) {
        if (tid < 32) {  // wave 0 issues the DMA (TDM ignores EXEC; one op/WG)
            const u64 ga = (u64)(img + (size_t)(y0 - HALO) * W_IMG + (x0 - HALO));
            const u32 lds_base = (u32)(size_t)(&tile[0]);
            const u32 tdim0 = (u32)(W_IMG - (x0 - HALO));  // remaining extent x
            const u32 tdim1 = (u32)(H_IMG - (y0 - HALO));  // remaining extent y

            // D# group 0 (128b): count=1 | lds_addr | global_addr | type=2
            u32x4 g0;
            g0[0] = 1u;                                  // count=1, user mode
            g0[1] = lds_base;                            // lds_addr (bytes)
            g0[2] = (u32)ga;                             // global_addr[31:0]
            g0[3] = (u32)((ga >> 32) & 0x1FFFFFFu)       // global_addr[56:32]
                    | (2u << 30);                        // type=2 ("image")

            // D# group 1 (256b): mask=0, data_size=4B, dims/strides, tile 68x20
            u32x8 g1;
            g1[0] = 2u << 16;                            // data_size=2 (4 bytes)
            g1[1] = (tdim0 & 0xFFFFu) << 16;             // tensor_dim0[15:0]
            g1[2] = (tdim0 >> 16) | ((tdim1 & 0xFFFFu) << 16);
            g1[3] = (tdim1 >> 16) | ((u32)LW << 16);     // tile_dim0 = 68
            g1[4] = (u32)LH;                             // tile_dim1 = 20, dim2=0
            g1[5] = (u32)W_IMG;                          // tensor_dim0_stride lo32
            g1[6] = 0u;                                  // stride0 hi, stride1 lo
            g1[7] = 0u;                                  // stride1 hi

            asm volatile("tensor_load_to_lds %0, %1"
                         :: "s"(g0), "s"(g1)
                         : "memory");
        }
        __builtin_amdgcn_s_wait_tensorcnt(0);   // no-op for non-issuing waves
        __syncthreads();
    } else {
        // Left/top border tiles: zero-init LDS (provides the zero padding),
        // then per-lane async copies of the in-bounds elements.
        for (int i = tid; i < LH * LW; i += NTHREADS)
            tile[i] = 0.0f;
        __syncthreads();

        for (int i = tid; i < LH * LW; i += NTHREADS) {
            const int r  = i / LW;
            const int c  = i - r * LW;
            const int gy = y0 - HALO + r;
            const int gx = x0 - HALO + c;
            if ((unsigned)gy < (unsigned)H_IMG && (unsigned)gx < (unsigned)W_IMG) {
                const u32    lds_addr = (u32)(size_t)(tile + i);
                const float* gp       = img + (size_t)gy * W_IMG + gx;
                asm volatile("global_load_async_to_lds_b32 %0, %1, off"
                             :: "v"(lds_addr), "v"(gp)
                             : "memory");
            }
        }
        asm volatile("s_wait_asynccnt 0" ::: "memory");
        __syncthreads();
    }

    // In-place ReLU over the staged tile (once per element, not per tap).
    for (int i = tid; i < LH * LW; i += NTHREADS)
        tile[i] = fmaxf(tile[i], 0.0f);
    __syncthreads();

    // Each thread computes 4 output pixels (64-wide rows -> coalesced I/O).
    const int lx  = tid & (TW - 1);
    const int ly0 = tid >> 6;          // 0..3
#pragma unroll
    for (int k = 0; k < 4; ++k) {
        const int ly = ly0 + 4 * k;    // 0..15
        const int gx = x0 + lx;
        const int gy = y0 + ly;

        // Per-pixel Gaussian bandwidth: w(r2) = t^r2, t = exp(-1/(2 s^2 + eps)).
        const float s  = scl[(size_t)gy * W_IMG + gx];
        const float a  = 1.0f / (2.0f * s * s + 1e-6f);
        const float t1 = __expf(-a);         // v_exp_f32 (hardware TRANS)
        const float t2 = t1 * t1;
        const float t4 = t2 * t2;
        const float t5 = t4 * t1;
        const float t8 = t4 * t4;

        // Group the 25 taps by squared radius r2 in {0,1,2,4,5,8}.
        const float c0 = T(0, 0);
        const float s1 = T(0, 1) + T(0, -1) + T(1, 0) + T(-1, 0);
        const float s2 = T(1, 1) + T(1, -1) + T(-1, 1) + T(-1, -1);
        const float s4 = T(0, 2) + T(0, -2) + T(2, 0) + T(-2, 0);
        const float s5 = T(1, 2) + T(1, -2) + T(-1, 2) + T(-1, -2)
                       + T(2, 1) + T(2, -1) + T(-2, 1) + T(-2, -1);
        const float s8 = T(2, 2) + T(2, -2) + T(-2, 2) + T(-2, -2);

        float conv = c0;
        conv = fmaf(s1, t1, conv);
        conv = fmaf(s2, t2, conv);
        conv = fmaf(s4, t4, conv);
        conv = fmaf(s5, t5, conv);
        conv = fmaf(s8, t8, conv);

        const size_t oidx = (size_t)n * NPIX + (size_t)gy * W_IMG + gx;
        out_conv[oidx] = conv;
        out_mask[oidx] = (conv >= 0.5f) ? 1.0f : 0.0f;
    }
}

extern "C" void kernel_launch(void* const* d_in, const int* in_sizes, int n_in,
                              void* d_out, int out_size, void* d_ws, size_t ws_size,
                              hipStream_t stream) {
    (void)n_in; (void)out_size; (void)d_ws; (void)ws_size;

    const float* bev_map   = (const float*)d_in[0];
    const float* bev_scale = (const float*)d_in[1];

    const int B = in_sizes[0] / NPIX;      // 4 in the reference setup

    float* out_conv = (float*)d_out;
    float* out_mask = out_conv + (size_t)B * NPIX;

    dim3 grid(W_IMG / TW, H_IMG / TH, B);  // 12 x 48 x B workgroups
    dim3 block(NTHREADS, 1, 1);            // 8 wave32s
    ConvThreshold_90108413870322_kernel<<<grid, block, 0, stream>>>(
        bev_map, bev_scale, out_conv, out_mask);
}